// MolecularGCN_77206332113764
// MI455X (gfx1250) — compile-verified
//
#include <hip/hip_runtime.h>
#include <hip/hip_bf16.h>
#include <math.h>

typedef __attribute__((ext_vector_type(2))) float v2f;
typedef __attribute__((ext_vector_type(8))) float v8f;

#define DD 128   // embedding dim, fixed by the model

// ---------------------------------------------------------------- utilities
__global__ void zero_f32(float* __restrict__ p, long long n) {
  long long i = (long long)blockIdx.x * blockDim.x + threadIdx.x;
  if (i < n) p[i] = 0.0f;
}

// deg starts at 1.0 (self-loop weight folded in here)
__global__ void deg_init(float* __restrict__ deg, int n) {
  int i = blockIdx.x * blockDim.x + threadIdx.x;
  if (i < n) deg[i] = 1.0f;
}

__global__ void deg_accum(const int* __restrict__ dst, const float* __restrict__ w,
                          float* __restrict__ deg, int E) {
  int e = blockIdx.x * blockDim.x + threadIdx.x;
  if (e < E) unsafeAtomicAdd(&deg[dst[e]], w[e]);   // native global_atomic_add_f32
}

// in-place deg -> d^-1/2
__global__ void deg_to_dinv(float* __restrict__ deg, int n) {
  int i = blockIdx.x * blockDim.x + threadIdx.x;
  if (i < n) {
    float d = deg[i];
    deg[i] = (d > 0.0f) ? rsqrtf(d) : 0.0f;
  }
}

// norm[e] = dinv[src] * w * dinv[dst]  (computed once, reused by all 3 layers)
__global__ void compute_norm(const int* __restrict__ src, const int* __restrict__ dst,
                             const float* __restrict__ w, const float* __restrict__ dinv,
                             float* __restrict__ norm, int E) {
  int e = blockIdx.x * blockDim.x + threadIdx.x;
  if (e < E) norm[e] = dinv[src[e]] * w[e] * dinv[dst[e]];
}

// ------------------------------------------------------- fp32 WMMA GEMM
// C[M x 128] = A[M x K] * B[K x 128].  One 16x16 output tile per wave;
// 8 waves per block cover all 128 columns of one 16-row strip.
// K is a compile-time constant: main loop fully unrolled with ZERO guards
// (immediate-offset loads), K%4 tail handled once, branchlessly (clamped
// address + v_cndmask select) so EXEC never changes around the WMMAs.
// Fragment layout per ISA 7.12.2:
//   A 16x4 f32: lanes 0-15 hold K={k,k+1}, lanes 16-31 hold K={k+2,k+3}, M=lane&15
//   B 4x16 mirrored with N on lanes; C/D: vgpr i -> row i+8*(lane>>4), col lane&15
template <int K, bool ALIGN8>
__global__ void gemm_wmma_f32(const float* __restrict__ A, const float* __restrict__ B,
                              float* __restrict__ C, int M) {
  const int lane = threadIdx.x & 31;
  const int wave = threadIdx.x >> 5;
  const int row0 = blockIdx.x << 4;      // 16-row strip
  const int col0 = wave << 4;            // 16-col tile (wave 0..7 -> 0..112)
  const int r    = lane & 15;
  const int kh   = (lane >> 4) << 1;     // 0 for lanes 0-15, 2 for lanes 16-31

  // branchless clamp keeps loads in-bounds without perturbing EXEC
  const int rowA = min(row0 + r, M - 1);
  const float* arow = A + (long long)rowA * K + kh;   // A[rowA, kh + ...]
  const float* bcol = B + (long long)kh * DD + col0 + r;

  constexpr int KMAIN = (K / 4) * 4;
  v8f acc = {};
#pragma unroll
  for (int k0 = 0; k0 < KMAIN; k0 += 4) {
    v2f a, b;
    if constexpr (ALIGN8) {
      a = *(const v2f*)(arow + k0);          // global_load_b64 (8B aligned)
    } else {
      a.x = arow[k0];
      a.y = arow[k0 + 1];
    }
    b.x = bcol[(long long)k0 * DD];          // immediate-offset b32 loads
    b.y = bcol[(long long)(k0 + 1) * DD];
    acc = __builtin_amdgcn_wmma_f32_16x16x4_f32(
        /*neg_a=*/false, a, /*neg_b=*/false, b,
        /*c_mod=*/(short)0, acc, /*reuse_a=*/false, /*reuse_b=*/false);
  }

  if constexpr (KMAIN < K) {                 // K % 4 tail (only the K=75 case)
    const int ka = KMAIN + kh;               // lane-dependent absolute k
    const int c0 = min(ka, K - 1);           // clamped, always in-bounds
    const int c1 = min(ka + 1, K - 1);
    const float av0 = A[(long long)rowA * K + c0];
    const float av1 = A[(long long)rowA * K + c1];
    const float bv0 = B[(long long)c0 * DD + col0 + r];
    const float bv1 = B[(long long)c1 * DD + col0 + r];
    v2f a, b;
    a.x = (ka     < K) ? av0 : 0.0f;         // v_cndmask, no EXEC branches
    a.y = (ka + 1 < K) ? av1 : 0.0f;
    b.x = (ka     < K) ? bv0 : 0.0f;
    b.y = (ka + 1 < K) ? bv1 : 0.0f;
    acc = __builtin_amdgcn_wmma_f32_16x16x4_f32(
        false, a, false, b, (short)0, acc, false, false);
  }

  // D layout: VGPR i -> row = row0 + i + 8*(lane>>4), col = col0 + (lane&15)
  const int mhi = (lane >> 4) << 3;
  float* crow = C + (long long)(row0 + mhi) * DD + col0 + r;
  if (row0 + 16 <= M) {                      // uniform (scalar) branch
#pragma unroll
    for (int i = 0; i < 8; ++i) crow[(long long)i * DD] = acc[i];
  } else {
#pragma unroll
    for (int i = 0; i < 8; ++i)
      if (row0 + mhi + i < M) crow[(long long)i * DD] = acc[i];
  }
}

// ------------------------------------------------- edge gather/scatter-add
// One wave per edge: 32 lanes x float4 = full 128-float row. xw and agg are
// L2-resident (25.6 MB each << 192 MB), so this runs at L2 atomic speed; one
// native f32 atomic per output float, no CAS loops.
__global__ void edge_scatter(const float* __restrict__ xw, const int* __restrict__ src,
                             const int* __restrict__ dst, const float* __restrict__ norm,
                             float* __restrict__ agg, int E) {
  const long long tid = (long long)blockIdx.x * blockDim.x + threadIdx.x;
  const int e    = (int)(tid >> 5);
  const int lane = threadIdx.x & 31;
  if (e >= E) return;
  const int   s  = src[e];
  const int   d  = dst[e];
  const float nm = norm[e];
  const float4 v = ((const float4*)(xw + (long long)s * DD))[lane];
  float* op = agg + (long long)d * DD + (lane << 2);
  unsafeAtomicAdd(op + 0, v.x * nm);
  unsafeAtomicAdd(op + 1, v.y * nm);
  unsafeAtomicAdd(op + 2, v.z * nm);
  unsafeAtomicAdd(op + 3, v.w * nm);
}

// ------------------------------------------ fused self-loop + bias + residual
// x += (agg + xw * dinv^2 + b[l]) * att[l]
__global__ void combine(float* __restrict__ x, const float* __restrict__ agg,
                        const float* __restrict__ xw, const float* __restrict__ dinv,
                        const float* __restrict__ b_layers, const float* __restrict__ att,
                        int layer, long long total) {
  long long i = (long long)blockIdx.x * blockDim.x + threadIdx.x;
  if (i >= total) return;
  const int row = (int)(i >> 7);   // /128
  const int col = (int)(i & 127);
  const float di  = dinv[row];
  const float out = agg[i] + xw[i] * (di * di) + b_layers[layer * DD + col];
  x[i] += out * att[layer];
}

// ---------------------------------------------------------------- launcher
extern "C" void kernel_launch(void* const* d_in, const int* in_sizes, int n_in,
                              void* d_out, int out_size, void* d_ws, size_t ws_size,
                              hipStream_t stream) {
  const float* embedding   = (const float*)d_in[0];
  const int*   edge_index  = (const int*)  d_in[1];
  const float* edge_weight = (const float*)d_in[2];
  const float* W_init      = (const float*)d_in[3];
  const float* W_layers    = (const float*)d_in[4];
  const float* b_layers    = (const float*)d_in[5];
  const float* att         = (const float*)d_in[6];

  const int IN_F = 75;               // compile-time K below must match
  const int N = in_sizes[0] / IN_F;
  const int E = in_sizes[2];
  const int L = in_sizes[5] / DD;

  const int* src = edge_index;       // edge_index[0, :]
  const int* dst = edge_index + E;   // edge_index[1, :]

  const long long ND = (long long)N * DD;
  float* ws   = (float*)d_ws;
  float* xw   = ws;                              // N*128
  float* agg  = xw + ND;                         // N*128
  float* dinv = agg + ND;                        // N (deg, then d^-1/2 in place)
  float* norm = dinv + (((long long)N + 63) & ~63LL);  // E

  float* x = (float*)d_out;                      // N*128, built in place

  const int T = 256;
  // --- GCN normalization (deg with self loops -> rsqrt -> per-edge norm)
  deg_init    <<<(N + T - 1) / T, T, 0, stream>>>(dinv, N);
  deg_accum   <<<(E + T - 1) / T, T, 0, stream>>>(dst, edge_weight, dinv, E);
  deg_to_dinv <<<(N + T - 1) / T, T, 0, stream>>>(dinv, N);
  compute_norm<<<(E + T - 1) / T, T, 0, stream>>>(src, dst, edge_weight, dinv, norm, E);

  // --- init transform: x = embedding @ W_init   (K = 75, branchless tail)
  gemm_wmma_f32<75, false><<<(N + 15) / 16, T, 0, stream>>>(embedding, W_init, x, N);

  // --- 3 GCN layers
  for (int l = 0; l < L; ++l) {
    gemm_wmma_f32<128, true><<<(N + 15) / 16, T, 0, stream>>>(
        x, W_layers + (long long)l * DD * DD, xw, N);
    zero_f32<<<(int)((ND + T - 1) / T), T, 0, stream>>>(agg, ND);
    const long long sthreads = (long long)E * 32;          // one wave per edge
    edge_scatter<<<(int)((sthreads + T - 1) / T), T, 0, stream>>>(
        xw, src, dst, norm, agg, E);
    combine<<<(int)((ND + T - 1) / T), T, 0, stream>>>(
        x, agg, xw, dinv, b_layers, att, l, ND);
  }
}